// GAT_48533130445251
// MI455X (gfx1250) — compile-verified
//
#include <hip/hip_runtime.h>
#include <cstdint>

typedef __attribute__((ext_vector_type(2))) float v2f;
typedef __attribute__((ext_vector_type(8))) float v8f;

#define NEG_SLOPE 0.2f
// ford(-inf) = ~bits(-inf) = ~0xFF800000 = 0x007FFFFF
#define ORDER_NEG_INF 0x007FFFFFu

__device__ __forceinline__ unsigned ford(float f) {
  unsigned u = __float_as_uint(f);
  return (u >> 31) ? ~u : (u | 0x80000000u);
}
__device__ __forceinline__ float funo(unsigned u) {
  return (u >> 31) ? __uint_as_float(u & 0x7FFFFFFFu) : __uint_as_float(~u);
}
__device__ __forceinline__ float lrelu(float v) {
  return v > 0.f ? v : NEG_SLOPE * v;
}

// ---------------------------------------------------------------------------
// WMMA fp32 GEMM: Y[N, COLS] = X[N, 128] @ W[128, COLS]
// One wave per 16-row tile, all COLS/16 column tiles held in accumulators.
// W staged in LDS in PAIR-INTERLEAVED layout: sW[(k/2)][col][k%2], so each
// B fragment {W[k0][col], W[k0+1][col]} is one aligned ds_load_b64 straight
// into the consecutive even VGPR pair V_WMMA_F32_16X16X4_F32 expects.
// ---------------------------------------------------------------------------
template <int COLS>
__global__ __launch_bounds__(128) void gemm_wmma_f32(
    const float* __restrict__ X, const float* __restrict__ W,
    float* __restrict__ Y, int nRowTiles) {
  __shared__ float sW[128 * COLS];
  const int tid = threadIdx.x;
  // cooperative staging: W row-major [128][COLS] -> pair-interleaved LDS
  for (int i = tid * 4; i < 128 * COLS; i += 128 * 4) {
    float4 w = *(const float4*)(W + i);
    const int k = i / COLS;  // row (i, COLS multiples of 4: no row crossing)
    const int c = i % COLS;  // first col
    float* dst = sW + (k >> 1) * (COLS * 2) + (k & 1);
    dst[(c + 0) * 2] = w.x;
    dst[(c + 1) * 2] = w.y;
    dst[(c + 2) * 2] = w.z;
    dst[(c + 3) * 2] = w.w;
  }
  __syncthreads();

  const int wave = tid >> 5;
  const int lane = tid & 31;
  const int rt = blockIdx.x * 4 + wave;
  if (rt >= nRowTiles) return;  // wave-uniform: EXEC stays all-ones

  const int half = lane >> 4;   // 0: K+0/1 ; 1: K+2/3 (A rows / B rows)
  const int lm = lane & 15;     // M for A-frag, N for B-frag / C
  const int rowbase = rt * 16;
  constexpr int NT = COLS / 16;

  v8f acc[NT] = {};
  const float* xrow = X + (size_t)(rowbase + lm) * 128;

  for (int k = 0; k < 128; k += 4) {
    const int k0 = k + half * 2;  // even
    float2 av = *(const float2*)(xrow + k0);
    v2f a;
    a.x = av.x;  // A[M=lm][k0]
    a.y = av.y;  // A[M=lm][k0+1]
    const float* brow = sW + (k0 >> 1) * (COLS * 2);
#pragma unroll
    for (int j = 0; j < NT; ++j) {
      v2f b = *(const v2f*)(brow + (j * 16 + lm) * 2);  // one ds_load_b64
      acc[j] = __builtin_amdgcn_wmma_f32_16x16x4_f32(
          false, a, false, b, (short)0, acc[j], false, false);
    }
  }

  // C/D layout: lane lm holds column N=lm; VGPR r holds row M = half*8 + r
#pragma unroll
  for (int j = 0; j < NT; ++j) {
#pragma unroll
    for (int r = 0; r < 8; ++r) {
      const int row = rowbase + half * 8 + r;
      Y[(size_t)row * COLS + j * 16 + lm] = acc[j][r];
    }
  }
}

// ---------------------------------------------------------------------------
// Layer 1: alpha dots per (node, head) + init emax/denom/agg
// ---------------------------------------------------------------------------
__global__ void alpha1_init(const float* __restrict__ h1,
                            const float* __restrict__ a_src,
                            const float* __restrict__ a_dst,
                            float* __restrict__ as, float* __restrict__ ad,
                            unsigned* __restrict__ emax,
                            float* __restrict__ denom, float* __restrict__ agg,
                            int N) {
  int i = blockIdx.x * blockDim.x + threadIdx.x;
  if (i >= N * 8) return;
  int n = i >> 3, hd = i & 7;
  const float* hp = h1 + (size_t)n * 128 + hd * 16;
  float s = 0.f, d = 0.f;
#pragma unroll
  for (int c = 0; c < 16; ++c) {
    float v = hp[c];
    s += v * a_src[hd * 16 + c];
    d += v * a_dst[hd * 16 + c];
  }
  as[i] = s;
  ad[i] = d;
  emax[i] = ORDER_NEG_INF;
  denom[i] = 0.f;
  float* ap = agg + (size_t)n * 128 + hd * 16;
#pragma unroll
  for (int c = 0; c < 16; ++c) ap[c] = 0.f;
}

// ---------------------------------------------------------------------------
// Edge pass A: segment max via ordered-uint atomicMax. One thread per (e,head)
// Edges e >= E are self-loops (src=dst=e-E).
// ---------------------------------------------------------------------------
template <int H>
__global__ void edge_pass_a(const int* __restrict__ ei, int E, int N,
                            const float* __restrict__ as,
                            const float* __restrict__ ad,
                            unsigned* __restrict__ emax) {
  int t = blockIdx.x * blockDim.x + threadIdx.x;
  int total = (E + N) * H;
  if (t >= total) return;
  int hd = t % H;
  int e = t / H;
  int s, d;
  if (e < E) { s = ei[e]; d = ei[E + e]; } else { s = d = e - E; }
  float v = lrelu(as[s * H + hd] + ad[d * H + hd]);
  atomicMax(&emax[d * H + hd], ford(v));
}

// ---------------------------------------------------------------------------
// Edge pass B: ex = exp(e - emax[dst]); cache ex; atomic-sum into denom
// ---------------------------------------------------------------------------
template <int H>
__global__ void edge_pass_b(const int* __restrict__ ei, int E, int N,
                            const float* __restrict__ as,
                            const float* __restrict__ ad,
                            const unsigned* __restrict__ emax,
                            float* __restrict__ denom, float* __restrict__ ex) {
  int t = blockIdx.x * blockDim.x + threadIdx.x;
  int total = (E + N) * H;
  if (t >= total) return;
  int hd = t % H;
  int e = t / H;
  int s, d;
  if (e < E) { s = ei[e]; d = ei[E + e]; } else { s = d = e - E; }
  float v = lrelu(as[s * H + hd] + ad[d * H + hd]);
  float x = __expf(v - funo(emax[d * H + hd]));
  ex[t] = x;
  atomicAdd(&denom[d * H + hd], x);
}

// ---------------------------------------------------------------------------
// Edge pass C, layer 1: one thread per (edge, head), 16 channels scatter-add
// ---------------------------------------------------------------------------
__global__ void edge_pass_c1(const int* __restrict__ ei, int E, int N,
                             const float* __restrict__ h1,
                             const float* __restrict__ ex,
                             const float* __restrict__ denom,
                             float* __restrict__ agg) {
  int t = blockIdx.x * blockDim.x + threadIdx.x;
  int total = (E + N) * 8;
  if (t >= total) return;
  int hd = t & 7;
  int e = t >> 3;
  int s, d;
  if (e < E) { s = ei[e]; d = ei[E + e]; } else { s = d = e - E; }
  float w = ex[t] / denom[d * 8 + hd];
  const float4* hp = (const float4*)(h1 + (size_t)s * 128 + hd * 16);
  float* ap = agg + (size_t)d * 128 + hd * 16;
#pragma unroll
  for (int q = 0; q < 4; ++q) {
    float4 hv = hp[q];
    atomicAdd(ap + q * 4 + 0, hv.x * w);
    atomicAdd(ap + q * 4 + 1, hv.y * w);
    atomicAdd(ap + q * 4 + 2, hv.z * w);
    atomicAdd(ap + q * 4 + 3, hv.w * w);
  }
}

// ---------------------------------------------------------------------------
// bias + ELU -> layer 2 input
// ---------------------------------------------------------------------------
__global__ void finalize1(const float* __restrict__ agg,
                          const float* __restrict__ b1,
                          float* __restrict__ h2in, int total) {
  int i = blockIdx.x * blockDim.x + threadIdx.x;
  if (i >= total) return;
  float v = agg[i] + b1[i & 127];
  h2in[i] = v > 0.f ? v : (__expf(v) - 1.f);
}

// ---------------------------------------------------------------------------
// Layer 2: alpha dots (H=1, C=64) + init
// ---------------------------------------------------------------------------
__global__ void alpha2_init(const float* __restrict__ h2,
                            const float* __restrict__ a_src,
                            const float* __restrict__ a_dst,
                            float* __restrict__ as, float* __restrict__ ad,
                            unsigned* __restrict__ emax,
                            float* __restrict__ denom, int N) {
  int n = blockIdx.x * blockDim.x + threadIdx.x;
  if (n >= N) return;
  const float4* hp = (const float4*)(h2 + (size_t)n * 64);
  float s = 0.f, d = 0.f;
#pragma unroll
  for (int q = 0; q < 16; ++q) {
    float4 hv = hp[q];
    s += hv.x * a_src[q * 4 + 0] + hv.y * a_src[q * 4 + 1] +
         hv.z * a_src[q * 4 + 2] + hv.w * a_src[q * 4 + 3];
    d += hv.x * a_dst[q * 4 + 0] + hv.y * a_dst[q * 4 + 1] +
         hv.z * a_dst[q * 4 + 2] + hv.w * a_dst[q * 4 + 3];
  }
  as[n] = s;
  ad[n] = d;
  emax[n] = ORDER_NEG_INF;
  denom[n] = 0.f;
}

__global__ void zero_f32(float* __restrict__ p, int total) {
  int i = blockIdx.x * blockDim.x + threadIdx.x;
  if (i < total) p[i] = 0.f;
}

// ---------------------------------------------------------------------------
// Edge pass C, layer 2: one thread per (edge, quarter), 16 channels each,
// scatter-add directly into d_out (pre-zeroed).
// ---------------------------------------------------------------------------
__global__ void edge_pass_c2(const int* __restrict__ ei, int E, int N,
                             const float* __restrict__ h2,
                             const float* __restrict__ ex,
                             const float* __restrict__ denom,
                             float* __restrict__ out) {
  int t = blockIdx.x * blockDim.x + threadIdx.x;
  int total = (E + N) * 4;
  if (t >= total) return;
  int q = t & 3;
  int e = t >> 2;
  int s, d;
  if (e < E) { s = ei[e]; d = ei[E + e]; } else { s = d = e - E; }
  float w = ex[e] / denom[d];
  const float4* hp = (const float4*)(h2 + (size_t)s * 64 + q * 16);
  float* op = out + (size_t)d * 64 + q * 16;
#pragma unroll
  for (int i = 0; i < 4; ++i) {
    float4 hv = hp[i];
    atomicAdd(op + i * 4 + 0, hv.x * w);
    atomicAdd(op + i * 4 + 1, hv.y * w);
    atomicAdd(op + i * 4 + 2, hv.z * w);
    atomicAdd(op + i * 4 + 3, hv.w * w);
  }
}

__global__ void add_bias2(float* __restrict__ out, const float* __restrict__ b2,
                          int total) {
  int i = blockIdx.x * blockDim.x + threadIdx.x;
  if (i >= total) return;
  out[i] = out[i] + b2[i & 63];
}

// ---------------------------------------------------------------------------
extern "C" void kernel_launch(void* const* d_in, const int* in_sizes, int n_in,
                              void* d_out, int out_size, void* d_ws,
                              size_t ws_size, hipStream_t stream) {
  (void)n_in; (void)out_size; (void)ws_size;
  const float* x      = (const float*)d_in[0];
  const int*   ei     = (const int*)d_in[1];
  const float* W1     = (const float*)d_in[2];
  const float* a1_src = (const float*)d_in[3];
  const float* a1_dst = (const float*)d_in[4];
  const float* b1     = (const float*)d_in[5];
  const float* W2     = (const float*)d_in[6];
  const float* a2_src = (const float*)d_in[7];
  const float* a2_dst = (const float*)d_in[8];
  const float* b2     = (const float*)d_in[9];
  float* out = (float*)d_out;

  const int N = in_sizes[0] / 128;   // 100000
  const int E = in_sizes[1] / 2;     // 1600000
  const int EE = E + N;              // with self-loops

  // workspace carving (floats); regions aliased across phases
  float* wsf = (float*)d_ws;
  size_t p = 0;
  float* h1   = wsf + p; p += (size_t)N * 128;  // later: h2in (ELU output)
  float* agg1 = wsf + p; p += (size_t)N * 128;  // later: h2 (layer-2 features)
  float* ex1  = wsf + p; p += (size_t)EE * 8;   // later: ex2 (first EE floats)
  float* as1  = wsf + p; p += (size_t)N * 8;
  float* ad1  = wsf + p; p += (size_t)N * 8;
  unsigned* emax1 = (unsigned*)(wsf + p); p += (size_t)N * 8;
  float* denom1 = wsf + p; p += (size_t)N * 8;
  float* as2  = wsf + p; p += N;
  float* ad2  = wsf + p; p += N;
  unsigned* emax2 = (unsigned*)(wsf + p); p += N;
  float* denom2 = wsf + p; p += N;

  const int TB = 256;
  const int rowTiles = N / 16;  // N is a multiple of 16 for this workload

  // ---- Layer 1 ----
  gemm_wmma_f32<128><<<(rowTiles + 3) / 4, 128, 0, stream>>>(x, W1, h1, rowTiles);
  alpha1_init<<<(N * 8 + TB - 1) / TB, TB, 0, stream>>>(
      h1, a1_src, a1_dst, as1, ad1, emax1, denom1, agg1, N);
  edge_pass_a<8><<<(EE * 8 + TB - 1) / TB, TB, 0, stream>>>(ei, E, N, as1, ad1, emax1);
  edge_pass_b<8><<<(EE * 8 + TB - 1) / TB, TB, 0, stream>>>(
      ei, E, N, as1, ad1, emax1, denom1, ex1);
  edge_pass_c1<<<(EE * 8 + TB - 1) / TB, TB, 0, stream>>>(
      ei, E, N, h1, ex1, denom1, agg1);
  finalize1<<<(N * 128 + TB - 1) / TB, TB, 0, stream>>>(agg1, b1, h1, N * 128);

  // ---- Layer 2 (h2in aliases h1, h2 aliases agg1, ex2 aliases ex1) ----
  float* h2in = h1;
  float* h2 = agg1;
  float* ex2 = ex1;
  gemm_wmma_f32<64><<<(rowTiles + 3) / 4, 128, 0, stream>>>(h2in, W2, h2, rowTiles);
  zero_f32<<<(N * 64 + TB - 1) / TB, TB, 0, stream>>>(out, N * 64);
  alpha2_init<<<(N + TB - 1) / TB, TB, 0, stream>>>(
      h2, a2_src, a2_dst, as2, ad2, emax2, denom2, N);
  edge_pass_a<1><<<(EE + TB - 1) / TB, TB, 0, stream>>>(ei, E, N, as2, ad2, emax2);
  edge_pass_b<1><<<(EE + TB - 1) / TB, TB, 0, stream>>>(
      ei, E, N, as2, ad2, emax2, denom2, ex2);
  edge_pass_c2<<<(EE * 4 + TB - 1) / TB, TB, 0, stream>>>(
      ei, E, N, h2, ex2, denom2, out);
  add_bias2<<<(N * 64 + TB - 1) / TB, TB, 0, stream>>>(out, b2, N * 64);
}